// ContrastiveLoss_19258633355799
// MI455X (gfx1250) — compile-verified
//
#include <hip/hip_runtime.h>
#include <hip/hip_bf16.h>
#include <math.h>

typedef __attribute__((ext_vector_type(2))) float v2f;
typedef __attribute__((ext_vector_type(8))) float v8f;

#define NB   4096          // batch
#define NROW 8192          // 2B
#define DIM  128
#define NT   512           // 16x16 tiles per side

// ---------------------------------------------------------------------------
// Kernel 1: L2-normalize rows of proj_1/proj_2 into reps[8192][128] (f32),
// zero the per-row denominator accumulator and the scalar output.
// One wave32 per row; lane l holds 4 contiguous floats.
// ---------------------------------------------------------------------------
__global__ __launch_bounds__(256) void nrm_kernel(const float* __restrict__ p1,
                                                  const float* __restrict__ p2,
                                                  float* __restrict__ reps,
                                                  float* __restrict__ rowsum,
                                                  float* __restrict__ out) {
    if (blockIdx.x == 0 && threadIdx.x == 0) out[0] = 0.0f;
    const int row  = (blockIdx.x * blockDim.x + threadIdx.x) >> 5;
    const int lane = threadIdx.x & 31;
    if (row >= NROW) return;
    const float* src = (row < NB) ? (p1 + (size_t)row * DIM)
                                  : (p2 + (size_t)(row - NB) * DIM);
    float4 v = ((const float4*)src)[lane];
    float ss = v.x * v.x + v.y * v.y + v.z * v.z + v.w * v.w;
    #pragma unroll
    for (int m = 16; m >= 1; m >>= 1) ss += __shfl_xor(ss, m, 32);
    const float scale = 1.0f / fmaxf(sqrtf(ss), 1e-12f);
    float4 o;
    o.x = v.x * scale; o.y = v.y * scale; o.z = v.z * scale; o.w = v.w * scale;
    ((float4*)(reps + (size_t)row * DIM))[lane] = o;
    if (lane == 0) rowsum[row] = 0.0f;
}

// ---------------------------------------------------------------------------
// Kernel 2: fused sim-tile + exp + row-sum using V_WMMA_F32_16X16X4_F32.
// Block = 8 waves = one 16-row tile; wave w sweeps column tiles w, w+8, ...
// A block (16x128 f32) kept in 64 VGPRs/lane; per column tile: 32 B64 loads
// + 32 WMMAs (K=128), then exp(2*sim) with diagonal masked out, accumulated
// locally and flushed with one atomic per row at the end.
// ---------------------------------------------------------------------------
__global__ __launch_bounds__(256) void sim_kernel(const float* __restrict__ reps,
                                                  float* __restrict__ rowsum) {
    const int lane = threadIdx.x & 31;
    const int wave = threadIdx.x >> 5;     // 0..7
    const int rt   = blockIdx.x;           // 0..511 row tile
    const int l15  = lane & 15;
    const int hi   = lane >> 4;            // 0: K={0,1}; 1: K={2,3} per k-step

    // Preload A: lane l -> row rt*16+l15, pairs (4kk+2hi, 4kk+2hi+1)
    const v2f* ap = (const v2f*)(reps + (size_t)(rt * 16 + l15) * DIM);
    v2f A[32];
    #pragma unroll
    for (int kk = 0; kk < 32; ++kk) A[kk] = ap[2 * kk + hi];

    float denom[8];
    #pragma unroll
    for (int v = 0; v < 8; ++v) denom[v] = 0.0f;

    for (int t = 0; t < NT / 8; ++t) {
        const int ct  = wave + (t << 3);   // column tile 0..511
        const int col = ct * 16 + l15;
        const v2f* bp = (const v2f*)(reps + (size_t)col * DIM);
        v8f acc = {0.f, 0.f, 0.f, 0.f, 0.f, 0.f, 0.f, 0.f};
        #pragma unroll
        for (int kk = 0; kk < 32; ++kk) {
            v2f b = bp[2 * kk + hi];
            acc = __builtin_amdgcn_wmma_f32_16x16x4_f32(
                false, A[kk], false, b, (short)0, acc, false, false);
        }
        // C layout: VGPR v, lanes 0-15 -> M=v, lanes 16-31 -> M=v+8; N=l15
        #pragma unroll
        for (int v = 0; v < 8; ++v) {
            const int rw = rt * 16 + v + 8 * hi;
            const float e = __expf(2.0f * acc[v]);   // sim / T, T = 0.5
            denom[v] += (rw == col) ? 0.0f : e;      // mask = 1 - eye
        }
    }
    // Row sums live across 16 lanes of each half -> xor-reduce within halves.
    #pragma unroll
    for (int v = 0; v < 8; ++v) {
        float s = denom[v];
        s += __shfl_xor(s, 1, 32);
        s += __shfl_xor(s, 2, 32);
        s += __shfl_xor(s, 4, 32);
        s += __shfl_xor(s, 8, 32);
        denom[v] = s;
    }
    if (l15 < 8)
        atomicAdd(&rowsum[rt * 16 + l15 + 8 * hi], denom[l15]);
}

// ---------------------------------------------------------------------------
// Kernel 3: per row r, pos = dot(reps[r], reps[partner]); accumulate
// (log(denom_r) - 2*pos) / N into out[0]. One wave per row.
// ---------------------------------------------------------------------------
__global__ __launch_bounds__(256) void fin_kernel(const float* __restrict__ reps,
                                                  const float* __restrict__ rowsum,
                                                  float* __restrict__ out) {
    const int row  = (blockIdx.x * blockDim.x + threadIdx.x) >> 5;
    const int lane = threadIdx.x & 31;
    if (row >= NROW) return;
    const int prt = (row < NB) ? row + NB : row - NB;
    float4 a = ((const float4*)(reps + (size_t)row * DIM))[lane];
    float4 b = ((const float4*)(reps + (size_t)prt * DIM))[lane];
    float dp = a.x * b.x + a.y * b.y + a.z * b.z + a.w * b.w;
    #pragma unroll
    for (int m = 16; m >= 1; m >>= 1) dp += __shfl_xor(dp, m, 32);
    if (lane == 0) {
        const float term = logf(rowsum[row]) - 2.0f * dp;
        atomicAdd(out, term * (1.0f / (float)NROW));
    }
}

extern "C" void kernel_launch(void* const* d_in, const int* in_sizes, int n_in,
                              void* d_out, int out_size, void* d_ws, size_t ws_size,
                              hipStream_t stream) {
    const float* p1 = (const float*)d_in[0];
    const float* p2 = (const float*)d_in[1];
    float* out = (float*)d_out;

    float* reps   = (float*)d_ws;                       // 8192*128 f32 = 4 MB
    float* rowsum = reps + (size_t)NROW * DIM;          // 8192 f32

    // 1 wave per row, 8 waves per block -> 1024 blocks
    nrm_kernel<<<NROW / 8, 256, 0, stream>>>(p1, p2, reps, rowsum, out);
    // 1 block per 16-row tile
    sim_kernel<<<NT, 256, 0, stream>>>(reps, rowsum);
    // 1 wave per row
    fin_kernel<<<NROW / 8, 256, 0, stream>>>(reps, rowsum, out);
}